// MixturePrior_63041529970783
// MI455X (gfx1250) — compile-verified
//
#include <hip/hip_runtime.h>

// Problem constants (from reference):
//   N = 32768 points, K = 2048 components, Z_DIM = 64
#define NROWS   32768
#define KCOMP   2048
#define ZDIM    64
// 64*(0.5*ln(2*pi) + 0.5*ln(64)) = 32*ln(128*pi)
#define LOGNORM 191.8963247926085f
#define LOG2E   1.4426950408889634f
#define LN2     0.6931471805599453f
#define L2E64   0.022542110013890054f   // LOG2E / 64

typedef __attribute__((ext_vector_type(2))) float v2f;
typedef __attribute__((ext_vector_type(8))) float v8f;

// ---------------------------------------------------------------------------
// Prologue: colconst2[k] = (logits[k] - logsumexp(logits) - |locs_k|^2/128)
//                          * LOG2E        (base-2 domain for v_exp_f32)
// ---------------------------------------------------------------------------
__global__ __launch_bounds__(256) void mp_colconst(const float* __restrict__ locs,
                                                   const float* __restrict__ logits,
                                                   float* __restrict__ colconst2) {
  __shared__ float red[256];
  const int t = threadIdx.x;

  float mx = -__builtin_inff();
  for (int k = t; k < KCOMP; k += 256) mx = fmaxf(mx, logits[k]);
  red[t] = mx;
  __syncthreads();
  for (int o = 128; o > 0; o >>= 1) {
    if (t < o) red[t] = fmaxf(red[t], red[t + o]);
    __syncthreads();
  }
  const float gmax = red[0];
  __syncthreads();

  float se = 0.f;
  for (int k = t; k < KCOMP; k += 256) se += __expf(logits[k] - gmax);
  red[t] = se;
  __syncthreads();
  for (int o = 128; o > 0; o >>= 1) {
    if (t < o) red[t] += red[t + o];
    __syncthreads();
  }
  const float lse = gmax + __logf(red[0]);

  for (int k = t; k < KCOMP; k += 256) {
    const float4* lr = (const float4*)(locs + (size_t)k * ZDIM);
    float ln2v = 0.f;
#pragma unroll
    for (int i = 0; i < 16; ++i) {
      float4 q = lr[i];
      ln2v += q.x * q.x + q.y * q.y + q.z * q.z + q.w * q.w;
    }
    colconst2[k] = (logits[k] - lse - ln2v * (1.0f / 128.0f)) * LOG2E;
  }
}

// ---------------------------------------------------------------------------
// Main kernel: 128 threads = 4 waves; wave w owns rows row0..row0+15 of x
// (A tile in registers). Double-buffered ASYNC global->LDS staging of 16-
// component locs tiles (global_load_async_to_lds_b128 + s_wait_asynccnt),
// V_WMMA_F32_16X16X4_F32 x16 per tile, branchless base-2 online
// logsumexp/argmax, 16-lane butterfly merge.
// ---------------------------------------------------------------------------
__global__ __launch_bounds__(128) void mp_main(const float* __restrict__ x,
                                               const float* __restrict__ locs,
                                               const float* __restrict__ colconst2,
                                               float* __restrict__ log_prob,
                                               float* __restrict__ quantized) {
  __shared__ float lB[2][16][68];   // double buffer; pad 4 -> conflict-free reads
  __shared__ float lxn2[4][16];

  const int tid  = threadIdx.x;
  const int lane = tid & 31;
  const int wave = tid >> 5;
  const int half = lane >> 4;       // 32-bit A layout: half selects K-pair
  const int lm   = lane & 15;
  const int row0 = (blockIdx.x * 4 + wave) * 16;

  // staging geometry (per thread): 8 floats = 2 x b128 of one locs row
  const int srow  = tid >> 3;
  const int scoff = (tid & 7) * 8;

  // ---- preload A fragments: x[row0+lm][*], lane holds k = 4s+2*half, +1 ----
  v2f afrag[16];
  const float* xrow = x + (size_t)(row0 + lm) * ZDIM;
  float psum = 0.f;
#pragma unroll
  for (int s = 0; s < 16; ++s) {
    const int kb = 4 * s + 2 * half;
    float2 av = *(const float2*)(xrow + kb);
    afrag[s].x = av.x;
    afrag[s].y = av.y;
    psum += av.x * av.x + av.y * av.y;
  }
  const float xn2 = psum + __shfl_xor(psum, 16, 32);   // |x_row|^2
  if (half == 0) lxn2[wave][lm] = xn2;

  float m_[8], s_[8];
  int arg_[8];
#pragma unroll
  for (int j = 0; j < 8; ++j) { m_[j] = -__builtin_inff(); s_[j] = 0.f; arg_[j] = 0; }

  // async stage of one 16x64 locs tile into LDS buffer b
#define STAGE(b, tile)                                                         \
  do {                                                                         \
    const float* g = locs + (size_t)((tile) * 16 + srow) * ZDIM + scoff;       \
    unsigned l = (unsigned)(unsigned long long)(uintptr_t)&lB[(b)][srow][scoff]; \
    asm volatile("global_load_async_to_lds_b128 %0, %2, off\n\t"               \
                 "global_load_async_to_lds_b128 %1, %3, off"                   \
                 :: "v"(l), "v"(l + 16u), "v"(g), "v"(g + 4)                   \
                 : "memory");                                                  \
  } while (0)

  STAGE(0, 0);
  asm volatile("s_wait_asynccnt 0x0" ::: "memory");
  __syncthreads();

  for (int tile = 0; tile < KCOMP / 16; ++tile) {
    const int cur = tile & 1;
    if (tile + 1 < KCOMP / 16) STAGE(cur ^ 1, tile + 1);   // prefetch next tile

    // ---- 16x16 tile of dot products: 16 x V_WMMA_F32_16X16X4_F32 ----
    v8f acc = {};
#pragma unroll
    for (int s = 0; s < 16; ++s) {
      const int kb = 4 * s + 2 * half;
      float2 bv = *(const float2*)&lB[cur][lm][kb];
      v2f b;
      b.x = bv.x;
      b.y = bv.y;
      acc = __builtin_amdgcn_wmma_f32_16x16x4_f32(false, afrag[s], false, b,
                                                  (short)0, acc, false, false);
    }

    // ---- branchless online logsumexp+argmax in base-2 domain ----
    const float cc2  = colconst2[tile * 16 + lm];
    const int   kcol = tile * 16 + lm;
#pragma unroll
    for (int j = 0; j < 8; ++j) {
      const float t2 = __builtin_fmaf(acc[j], L2E64, cc2); // (dot/64+cc)*log2e
      const float mo = fmaxf(m_[j], t2);
      s_[j] = __builtin_fmaf(s_[j], __builtin_amdgcn_exp2f(m_[j] - mo),
                             __builtin_amdgcn_exp2f(t2 - mo));
      arg_[j] = (t2 > m_[j]) ? kcol : arg_[j];             // strict >: first index
      m_[j] = mo;
    }

    asm volatile("s_wait_asynccnt 0x0" ::: "memory");      // next buffer landed
    __syncthreads();                                       // all waves done r/w
  }
#undef STAGE

  // ---- merge 16 column-lanes within each half (rows j+8*half) ----
#pragma unroll
  for (int j = 0; j < 8; ++j) {
    float m = m_[j], s = s_[j];
    int a = arg_[j];
#pragma unroll
    for (int mask = 1; mask <= 8; mask <<= 1) {
      const float mo = __shfl_xor(m, mask, 32);
      const float so = __shfl_xor(s, mask, 32);
      const int   ao = __shfl_xor(a, mask, 32);
      const bool take = (mo > m) || ((mo == m) && (ao < a));  // first-index ties
      const float mn  = fmaxf(m, mo);
      s = __builtin_fmaf(s, __builtin_amdgcn_exp2f(m - mn),
                         so * __builtin_amdgcn_exp2f(mo - mn));
      a = take ? ao : a;
      m = mn;
    }
    m_[j] = m; s_[j] = s; arg_[j] = a;
  }

  // ---- log_prob: row r = j + 8*half; convert base-2 lse back to natural ----
  if (lm == 0) {
#pragma unroll
    for (int j = 0; j < 8; ++j) {
      const int r = j + 8 * half;
      log_prob[row0 + r] = LN2 * (m_[j] + __builtin_amdgcn_logf(s_[j]))
                           - lxn2[wave][r] * (1.0f / 128.0f) - LOGNORM;
    }
  }

  // ---- quantized: arg_[j] is uniform within each half after the butterfly;
  //      half h copies row j+8h with 16 lanes x float4 (b128) ----
#pragma unroll
  for (int j = 0; j < 8; ++j) {
    const int r = j + 8 * half;
    const float4 v = ((const float4*)(locs + (size_t)arg_[j] * ZDIM))[lm];
    ((float4*)(quantized + (size_t)(row0 + r) * ZDIM))[lm] = v;
  }
}

// ---------------------------------------------------------------------------
extern "C" void kernel_launch(void* const* d_in, const int* in_sizes, int n_in,
                              void* d_out, int out_size, void* d_ws, size_t ws_size,
                              hipStream_t stream) {
  (void)in_sizes; (void)n_in; (void)out_size; (void)ws_size;
  const float* x      = (const float*)d_in[0];
  const float* locs   = (const float*)d_in[1];
  const float* logits = (const float*)d_in[2];

  float* out       = (float*)d_out;
  float* log_prob  = out;           // [N]
  float* quantized = out + NROWS;   // [N, 64]
  float* colconst2 = (float*)d_ws;  // [K] scratch (base-2 scaled)

  mp_colconst<<<dim3(1), dim3(256), 0, stream>>>(locs, logits, colconst2);
  mp_main<<<dim3(NROWS / 64), dim3(128), 0, stream>>>(x, locs, colconst2,
                                                      log_prob, quantized);
}